// FrequencyDomainPrivacyAdapter_65893388255346
// MI455X (gfx1250) — compile-verified
//
#include <hip/hip_runtime.h>
#include <math.h>

#define PI_F 3.14159265358979323846f
#define H 512
#define W 512
#define NIMG 96                         // 32 batch * 3 channels
#define IMG_PIX (512 * 512)             // 262144
#define MAXDIST 362.03867196751236f     // sqrt(256^2 + 256^2)

typedef float v2f __attribute__((ext_vector_type(2)));
typedef float v8f __attribute__((ext_vector_type(8)));

// ---------------------------------------------------------------------------
// f32 WMMA:  D(16x16) = A(16x4) * B(4x16) + C   (V_WMMA_F32_16X16X4_F32)
// ---------------------------------------------------------------------------
__device__ __forceinline__ v8f wmma_f32(v2f a, v2f b, v8f c) {
  return __builtin_amdgcn_wmma_f32_16x16x4_f32(
      false, a, false, b, (short)0, c, false, false);
}

// Complex matmul  D(16x32) = F(16x16) * S(16x32), all row-major in LDS.
// F is the DFT-16 matrix (Fr + i*Fi), S is (sre + i*sim).
// A-fragment layout (ISA 7.12.2, 32-bit A 16x4): lane<16 -> K={k0,k0+1},
// lane>=16 -> K={k0+2,k0+3}, M = lane&15.  B mirrors (rows across the VGPR
// pair, N = lane&15).  C/D: VGPR r -> M = r + 8*(lane>>4), N = lane&15.
__device__ __forceinline__ void cmatmul16x32(const float* __restrict__ Fr,
                                             const float* __restrict__ Fi,
                                             const float* sre, const float* sim,
                                             float* dre, float* dim_, int lane) {
  const int half = lane >> 4;
  const int l15  = lane & 15;
#pragma unroll
  for (int t = 0; t < 2; ++t) {
    v8f accR = {0.f, 0.f, 0.f, 0.f, 0.f, 0.f, 0.f, 0.f};
    v8f accI = {0.f, 0.f, 0.f, 0.f, 0.f, 0.f, 0.f, 0.f};
    const int col = t * 16 + l15;
#pragma unroll
    for (int k0 = 0; k0 < 16; k0 += 4) {
      const int ka = k0 + 2 * half;
      v2f aR, aI, aIN, bR, bI;
      aR[0]  = Fr[l15 * 16 + ka];      aR[1]  = Fr[l15 * 16 + ka + 1];
      aI[0]  = Fi[l15 * 16 + ka];      aI[1]  = Fi[l15 * 16 + ka + 1];
      aIN[0] = -aI[0];                 aIN[1] = -aI[1];
      bR[0]  = sre[ka * 32 + col];     bR[1]  = sre[(ka + 1) * 32 + col];
      bI[0]  = sim[ka * 32 + col];     bI[1]  = sim[(ka + 1) * 32 + col];
      // (Fr + iFi)(Sr + iSi):  Dr = Fr*Sr - Fi*Si ;  Di = Fr*Si + Fi*Sr
      accR = wmma_f32(aR,  bR, accR);
      accR = wmma_f32(aIN, bI, accR);
      accI = wmma_f32(aR,  bI, accI);
      accI = wmma_f32(aI,  bR, accI);
    }
#pragma unroll
    for (int r = 0; r < 8; ++r) {
      const int row = r + 8 * half;
      dre[row * 32 + col]  = accR[r];
      dim_[row * 32 + col] = accI[r];
    }
  }
}

// 512-pt FFT core (after radix-2 + W32 twiddle already deposited the 16x32
// matrix Z[m1][c] (c = 2*n1 + k2) into bufA):
//   bufB = F16 * bufA           (inner DFT-16, batched over 32 columns)
//   bufA = twiddle/reorg(bufB)  (W_512^{n1*k'},  [k1'][2n1+k2] -> [n1][2k1'+k2])
//   bufB = F16 * bufA           (outer DFT-16)   -> bufB[k1*32+k'] = X[32k1+k']
__device__ __forceinline__ void fft512_core(const float* Fr, const float* Fi,
                                            float* are, float* aim,
                                            float* bre, float* bim, int lane) {
  __syncthreads();
  cmatmul16x32(Fr, Fi, are, aim, bre, bim, lane);
  __syncthreads();
#pragma unroll
  for (int t = 0; t < 16; ++t) {
    const int e   = lane + 32 * t;
    const int k1p = e >> 5;
    const int c   = e & 31;
    const int n1  = c >> 1;
    const int kp  = 2 * k1p + (c & 1);
    const float ang = (2.0f * PI_F / 512.0f) * (float)(n1 * kp);
    const float s = __sinf(ang), cc = __cosf(ang);
    const float re = bre[e], im = bim[e];
    are[n1 * 32 + kp] = re * cc + im * s;   // * (cos - i sin)
    aim[n1 * 32 + kp] = im * cc - re * s;
  }
  __syncthreads();
  cmatmul16x32(Fr, Fi, are, aim, bre, bim, lane);
  __syncthreads();
}

__device__ __forceinline__ void init_dft16(float* Fr, float* Fi) {
  for (int t = threadIdx.x; t < 256; t += 128) {
    const int k = t >> 4, m = t & 15;
    const float ang = (2.0f * PI_F / 16.0f) * (float)(k * m);
    Fr[t] = __cosf(ang);
    Fi[t] = -__sinf(ang);     // W16^{km} = cos - i sin
  }
}

__device__ __forceinline__ int band_of(int y, int x) {
  const float dy = (float)y - 256.0f, dx = (float)x - 256.0f;
  const float dist = sqrtf(dy * dy + dx * dx);
  int b = (int)floorf(dist / MAXDIST * 8.0f);
  return b > 8 ? 8 : b;
}

// ---------------------------------------------------------------------------
// Pass 1: row FFTs of real input, transposed planar output interm[img][k][h]
// 128 threads = 4 waves = 4 rows per block.
// ---------------------------------------------------------------------------
__global__ void __launch_bounds__(128)
pass1_rowfft(const float* __restrict__ in, float* __restrict__ ore,
             float* __restrict__ oim) {
  __shared__ float Fr[256], Fi[256];
  __shared__ float wbuf[4 * 2048];
  const int wave = threadIdx.x >> 5;
  const int lane = threadIdx.x & 31;
  float* are = wbuf + wave * 2048;
  float* aim = are + 512;
  float* bre = are + 1024;
  float* bim = are + 1536;

  init_dft16(Fr, Fi);

  const long long g = (long long)blockIdx.x * 4 + wave;   // row id < 96*512
  const float* row = in + g * 512;
  // radix-2 over pairs (j, j+256) + W32^{m1} twiddle; real input (imag = 0)
#pragma unroll
  for (int t = 0; t < 8; ++t) {
    const int j  = lane + 32 * t;     // 0..255
    const int n1 = j & 15, m1 = j >> 4;
    const float u = row[j], v = row[j + 256];
    are[m1 * 32 + 2 * n1] = u + v;
    aim[m1 * 32 + 2 * n1] = 0.0f;
    const float d   = u - v;
    const float ang = (2.0f * PI_F / 32.0f) * (float)m1;
    are[m1 * 32 + 2 * n1 + 1] =  d * __cosf(ang);
    aim[m1 * 32 + 2 * n1 + 1] = -d * __sinf(ang);
  }

  fft512_core(Fr, Fi, are, aim, bre, bim, lane);

  // cooperative transposed write: interm[img][k][h]
  const long long g0  = (long long)blockIdx.x * 4;
  const long long img = g0 >> 9;
  const int h0 = (int)(g0 & 511);
  const int hl = threadIdx.x & 3;          // which wave's row
  const int kb = threadIdx.x >> 2;         // 0..31
  const float* sre = wbuf + hl * 2048 + 1024;
  const float* sim = sre + 512;
  float* pre = ore + img * (long long)IMG_PIX + h0 + hl;
  float* pim = oim + img * (long long)IMG_PIX + h0 + hl;
#pragma unroll
  for (int i = 0; i < 16; ++i) {
    const int k = kb + 32 * i;
    pre[(long long)k * 512] = sre[k];
    pim[(long long)k * 512] = sim[k];
  }
}

// ---------------------------------------------------------------------------
// Pass 2: column FFTs (contiguous reads of interm rows), magnitude + band
// sum / sum-of-squares / max accumulation. mag stored as [img][kh][kw].
// ---------------------------------------------------------------------------
__global__ void __launch_bounds__(128)
pass2_colfft(const float* __restrict__ ire, const float* __restrict__ iim,
             float* __restrict__ mag, float* __restrict__ gs1,
             float* __restrict__ gen, unsigned* __restrict__ gmx) {
  __shared__ float Fr[256], Fi[256];
  __shared__ float wbuf[4 * 2048];
  __shared__ float ls1[9], le[9];
  __shared__ unsigned lmx[9];
  const int wave = threadIdx.x >> 5;
  const int lane = threadIdx.x & 31;
  float* are = wbuf + wave * 2048;
  float* aim = are + 512;
  float* bre = are + 1024;
  float* bim = are + 1536;

  if (threadIdx.x < 9) { ls1[threadIdx.x] = 0.f; le[threadIdx.x] = 0.f; lmx[threadIdx.x] = 0u; }
  init_dft16(Fr, Fi);

  const long long g    = (long long)blockIdx.x * 4 + wave;  // (img, kw)
  const long long base = (g >> 9) * (long long)IMG_PIX + (g & 511) * 512ll;
  const float* rre = ire + base;
  const float* rim = iim + base;
#pragma unroll
  for (int t = 0; t < 8; ++t) {
    const int j  = lane + 32 * t;
    const int n1 = j & 15, m1 = j >> 4;
    const float ur = rre[j], ui = rim[j];
    const float vr = rre[j + 256], vi = rim[j + 256];
    are[m1 * 32 + 2 * n1] = ur + vr;
    aim[m1 * 32 + 2 * n1] = ui + vi;
    const float dr = ur - vr, di = ui - vi;
    const float ang = (2.0f * PI_F / 32.0f) * (float)m1;
    const float s = __sinf(ang), c = __cosf(ang);
    are[m1 * 32 + 2 * n1 + 1] = dr * c + di * s;
    aim[m1 * 32 + 2 * n1 + 1] = di * c - dr * s;
  }

  fft512_core(Fr, Fi, are, aim, bre, bim, lane);

  // magnitude into bufA (bufA is free after the core)
#pragma unroll
  for (int t = 0; t < 16; ++t) {
    const int k = lane + 32 * t;
    const float r = bre[k], q = bim[k];
    are[k] = sqrtf(r * r + q * q);
  }
  __syncthreads();

  // cooperative transposed mag write + band statistics
  const long long g0  = (long long)blockIdx.x * 4;
  const long long img = g0 >> 9;
  const int kw0 = (int)(g0 & 511);
  const int hl  = threadIdx.x & 3;
  const int kb  = threadIdx.x >> 2;
  const int kw  = kw0 + hl;
  const float* smag = wbuf + hl * 2048;
  float* pm = mag + img * (long long)IMG_PIX + kw;
#pragma unroll
  for (int i = 0; i < 16; ++i) {
    const int kh = kb + 32 * i;
    const float m = smag[kh];
    pm[(long long)kh * 512] = m;
    const int bin = band_of(kh, kw);
    atomicAdd(&ls1[bin], m);
    atomicAdd(&le[bin], m * m);
    atomicMax(&lmx[bin], __float_as_uint(m));   // m >= 0: bit-order == value-order
  }
  __syncthreads();
  if (threadIdx.x < 9) {
    atomicAdd(&gs1[threadIdx.x], ls1[threadIdx.x]);
    atomicAdd(&gen[threadIdx.x], le[threadIdx.x]);
    atomicMax(&gmx[threadIdx.x], lmx[threadIdx.x]);
  }
}

// ---------------------------------------------------------------------------
// Pass 3: sparsity = count(mag > 0.1 * band_max[bin]), bands 0..7 only.
// ---------------------------------------------------------------------------
__global__ void __launch_bounds__(256)
pass3_sparsity(const float* __restrict__ mag, const unsigned* __restrict__ gmx,
               float* __restrict__ gsp) {
  __shared__ float thr[8];
  __shared__ float cnt[8];
  if (threadIdx.x < 8) {
    thr[threadIdx.x] = 0.1f * __uint_as_float(gmx[threadIdx.x]);
    cnt[threadIdx.x] = 0.0f;
  }
  __syncthreads();
  const long long n = (long long)NIMG * IMG_PIX;
  for (long long idx = (long long)blockIdx.x * blockDim.x + threadIdx.x; idx < n;
       idx += (long long)gridDim.x * blockDim.x) {
    const int pix = (int)(idx & (IMG_PIX - 1));
    const int bin = band_of(pix >> 9, pix & 511);
    if (bin < 8 && mag[idx] > thr[bin]) atomicAdd(&cnt[bin], 1.0f);
  }
  __syncthreads();
  if (threadIdx.x < 8) atomicAdd(&gsp[threadIdx.x], cnt[threadIdx.x]);
}

// ---------------------------------------------------------------------------
// Pass 4: finalize -> out[8][5] = (mean, std, max, energy, sparsity)
// ---------------------------------------------------------------------------
__global__ void __launch_bounds__(256)
pass4_final(const float* __restrict__ gs1, const float* __restrict__ gen,
            const unsigned* __restrict__ gmx, const float* __restrict__ gsp,
            float* __restrict__ out) {
  __shared__ float cnt[9];
  if (threadIdx.x < 9) cnt[threadIdx.x] = 0.0f;
  __syncthreads();
  for (int pix = threadIdx.x; pix < IMG_PIX; pix += blockDim.x)
    atomicAdd(&cnt[band_of(pix >> 9, pix & 511)], 1.0f);
  __syncthreads();
  if (threadIdx.x < 8) {
    const int b = threadIdx.x;
    const float c = cnt[b];
    const float S = gs1[b], E = gen[b];
    const float mean  = S / c;
    const float n_all = (float)NIMG * c;
    const float var   = (E - 2.0f * mean * S + mean * mean * n_all) / c;
    out[b * 5 + 0] = mean;
    out[b * 5 + 1] = sqrtf(fmaxf(var, 0.0f));
    out[b * 5 + 2] = __uint_as_float(gmx[b]);
    out[b * 5 + 3] = E;
    out[b * 5 + 4] = gsp[b];
  }
}

__global__ void init_stats(float* gs1, float* gen, unsigned* gmx, float* gsp) {
  const int t = threadIdx.x;
  if (t < 9) { gs1[t] = 0.f; gen[t] = 0.f; gmx[t] = 0u; gsp[t] = 0.f; }
}

// ---------------------------------------------------------------------------
// Workspace layout (floats):
//   [0, NP)        interm_re   NP = 96*512*512 = 25,165,824
//   [NP, 2NP)      interm_im
//   [2NP, 3NP)     mag
//   [3NP, ...)     stats: s1[9] | energy[9] | maxbits[9] | sparsity[9]
// requires ws_size >= 3*NP*4 + 144 bytes (~302 MB)
// ---------------------------------------------------------------------------
extern "C" void kernel_launch(void* const* d_in, const int* in_sizes, int n_in,
                              void* d_out, int out_size, void* d_ws, size_t ws_size,
                              hipStream_t stream) {
  (void)in_sizes; (void)n_in; (void)out_size; (void)ws_size;
  const float* in = (const float*)d_in[0];
  float* out = (float*)d_out;

  const long long NP = (long long)NIMG * IMG_PIX;
  float* interm_re = (float*)d_ws;
  float* interm_im = interm_re + NP;
  float* magbuf    = interm_im + NP;
  float* gs1       = magbuf + NP;
  float* gen       = gs1 + 9;
  unsigned* gmx    = (unsigned*)(gen + 9);
  float* gsp       = (float*)(gmx + 9);

  const int nrows  = NIMG * 512;           // 49152
  const int blocks = nrows / 4;            // 4 waves (rows) per 128-thread block

  init_stats<<<1, 64, 0, stream>>>(gs1, gen, gmx, gsp);
  pass1_rowfft<<<blocks, 128, 0, stream>>>(in, interm_re, interm_im);
  pass2_colfft<<<blocks, 128, 0, stream>>>(interm_re, interm_im, magbuf, gs1, gen, gmx);
  pass3_sparsity<<<2048, 256, 0, stream>>>(magbuf, gmx, gsp);
  pass4_final<<<1, 256, 0, stream>>>(gs1, gen, gmx, gsp, out);
}